// GCN_37744172597774
// MI455X (gfx1250) — compile-verified
//
#include <hip/hip_runtime.h>

typedef __attribute__((ext_vector_type(2))) float v2f;
typedef __attribute__((ext_vector_type(8))) float v8f;

#define HID 32

// ---------------- degree / normalization ----------------

__global__ void k_deg_init(float* __restrict__ deg, int n) {
    int i = blockIdx.x * blockDim.x + threadIdx.x;
    if (i < n) deg[i] = 1.0f;                     // self-loop contributes 1
}

__global__ void k_deg_edges(const int* __restrict__ dst, float* __restrict__ deg, int e) {
    int i = blockIdx.x * blockDim.x + threadIdx.x;
    if (i < e) atomicAdd(&deg[dst[i]], 1.0f);
}

__global__ void k_dinv(float* __restrict__ deg, int n) {
    int i = blockIdx.x * blockDim.x + threadIdx.x;
    if (i < n) deg[i] = rsqrtf(deg[i]);           // deg >= 1 always (self loops)
}

// ---------------- dense GEMM via fp32 WMMA 16x16x4 ----------------
// H[n_rows, 32] = A[n_rows, K] @ W[K, 32]
// One wave computes a full 16x32 output tile (two 16x16 WMMA n-tiles sharing
// the A operand). A-rows are clamped (not predicated) so loads stay
// straight-line; only stores are range-guarded.
template <int K, bool RELU>
__global__ void __launch_bounds__(256) k_gemm_wmma(
    const float* __restrict__ A, const float* __restrict__ W,
    float* __restrict__ H, int n_rows)
{
    int wave  = (blockIdx.x * blockDim.x + threadIdx.x) >> 5;
    int lane  = threadIdx.x & 31;
    int n_mt  = (n_rows + 15) >> 4;
    int mtile = wave;
    if (mtile >= n_mt) return;             // wave-uniform exit: EXEC stays all-1s

    int m     = lane & 15;
    int kofs  = (lane >> 4) << 1;          // lanes 0-15 -> K+0/K+1, lanes 16-31 -> K+2/K+3
    int row   = mtile * 16 + m;
    int rowc  = row < n_rows ? row : (n_rows - 1);   // clamp: tail rows read valid data
    const v2f* arow = (const v2f*)(A + (size_t)rowc * K);
    int n0    = lane & 15;                 // B column within n-tile

    v8f acc0 = {};
    v8f acc1 = {};
    #pragma unroll
    for (int k0 = 0; k0 < K; k0 += 4) {
        v2f av = arow[(k0 + kofs) >> 1];   // one aligned b64 load
        if (RELU) { av.x = fmaxf(av.x, 0.0f); av.y = fmaxf(av.y, 0.0f); }
        const float* wr0 = W + (size_t)(k0 + kofs) * HID;
        v2f bv0, bv1;
        bv0.x = wr0[n0];        bv0.y = wr0[HID + n0];
        bv1.x = wr0[16 + n0];   bv1.y = wr0[HID + 16 + n0];
        acc0 = __builtin_amdgcn_wmma_f32_16x16x4_f32(
                   false, av, false, bv0, (short)0, acc0, false, false);
        acc1 = __builtin_amdgcn_wmma_f32_16x16x4_f32(
                   false, av, false, bv1, (short)0, acc1, false, false);
    }

    int mbase = mtile * 16 + ((lane >> 4) << 3);   // rows r / 8+r per lane half
    float* hp = H + (size_t)mbase * HID + n0;
    if (mbase + 8 <= n_rows) {                     // fast path: whole half-tile in range
        #pragma unroll
        for (int r = 0; r < 8; ++r) {
            hp[(size_t)r * HID]      = acc0[r];
            hp[(size_t)r * HID + 16] = acc1[r];
        }
    } else {
        #pragma unroll
        for (int r = 0; r < 8; ++r) {
            if (mbase + r < n_rows) {
                hp[(size_t)r * HID]      = acc0[r];
                hp[(size_t)r * HID + 16] = acc1[r];
            }
        }
    }
}

// ---------------- aggregation: self-loop + bias init, then edge scatter ----------------

__global__ void k_agg_init(const float* __restrict__ H, const float* __restrict__ dinv,
                           const float* __restrict__ b, float* __restrict__ Acc, int n)
{
    int gid = blockIdx.x * blockDim.x + threadIdx.x;
    int i = gid >> 5, c = gid & 31;
    if (i >= n) return;
    float di = dinv[i];
    Acc[(size_t)i * HID + c] = b[c] + H[(size_t)i * HID + c] * di * di;
}

__global__ void k_agg_edges(const int* __restrict__ src, const int* __restrict__ dst,
                            const float* __restrict__ dinv, const float* __restrict__ H,
                            float* __restrict__ Acc, int e)
{
    int gid = blockIdx.x * blockDim.x + threadIdx.x;
    int ed = gid >> 5, c = gid & 31;      // one wave per edge, one channel per lane
    if (ed >= e) return;
    int s = src[ed], d = dst[ed];
    float w = dinv[s] * dinv[d];
    float v = H[(size_t)s * HID + c] * w; // coalesced 128B gather per edge
    atomicAdd(&Acc[(size_t)d * HID + c], v);
}

// ---------------- pooling + FC ----------------

__global__ void k_pool_zero(float* __restrict__ P, float* __restrict__ cnt, int g) {
    int i = blockIdx.x * blockDim.x + threadIdx.x;
    if (i < g * HID) P[i] = 0.0f;
    if (i < g) cnt[i] = 0.0f;
}

__global__ void k_pool(const float* __restrict__ A3, const int* __restrict__ batch,
                       float* __restrict__ P, float* __restrict__ cnt, int n)
{
    int gid = blockIdx.x * blockDim.x + threadIdx.x;
    int i = gid >> 5, c = gid & 31;
    if (i >= n) return;
    int g = batch[i];
    atomicAdd(&P[(size_t)g * HID + c], A3[(size_t)i * HID + c]);
    if (c == 0) atomicAdd(&cnt[g], 1.0f);
}

__global__ void k_fc(const float* __restrict__ P, const float* __restrict__ cnt,
                     const float* __restrict__ Wfc, const float* __restrict__ bfc,
                     float* __restrict__ out, int g, int ncls)
{
    int gid = blockIdx.x * blockDim.x + threadIdx.x;
    if (gid >= g * ncls) return;
    int gr = gid / ncls, j = gid % ncls;
    float inv = 1.0f / fmaxf(cnt[gr], 1.0f);
    float acc = bfc[j];
    #pragma unroll
    for (int c = 0; c < HID; ++c)
        acc += P[(size_t)gr * HID + c] * inv * Wfc[c * ncls + j];
    out[gid] = acc;
}

// ---------------- launcher ----------------

extern "C" void kernel_launch(void* const* d_in, const int* in_sizes, int n_in,
                              void* d_out, int out_size, void* d_ws, size_t ws_size,
                              hipStream_t stream)
{
    const float* x    = (const float*)d_in[0];
    const int*   ei   = (const int*)d_in[1];
    const int*   bat  = (const int*)d_in[2];
    const float* W1   = (const float*)d_in[3];
    const float* b1   = (const float*)d_in[4];
    const float* W2   = (const float*)d_in[5];
    const float* b2   = (const float*)d_in[6];
    const float* W3   = (const float*)d_in[7];
    const float* b3   = (const float*)d_in[8];
    const float* Wfc  = (const float*)d_in[9];
    const float* bfc  = (const float*)d_in[10];
    float* out = (float*)d_out;

    const int FIN = 128, G = 256, NCLS = 10;
    int n = in_sizes[0] / FIN;
    int e = in_sizes[1] / 2;
    const int* srcI = ei;
    const int* dstI = ei + e;

    // workspace carve-out
    char* ws = (char*)d_ws;
    size_t off = 0;
    auto carve = [&](size_t bytes) {
        void* p = ws + off;
        off = (off + bytes + 255) & ~(size_t)255;
        return p;
    };
    float* dinv = (float*)carve((size_t)n * 4);
    float* Hbuf = (float*)carve((size_t)n * HID * 4);
    float* Abuf = (float*)carve((size_t)n * HID * 4);
    float* P    = (float*)carve((size_t)G * HID * 4);
    float* cnt  = (float*)carve((size_t)G * 4);

    const int T = 256;
    auto blk = [&](long long work) { return (unsigned)((work + T - 1) / T); };

    // normalization: deg -> dinv (in place)
    k_deg_init <<<blk(n), T, 0, stream>>>(dinv, n);
    k_deg_edges<<<blk(e), T, 0, stream>>>(dstI, dinv, e);
    k_dinv     <<<blk(n), T, 0, stream>>>(dinv, n);

    long long gemm_threads = (long long)((n + 15) / 16) * 32;  // one wave per 16-row tile
    long long nodeCh = (long long)n * HID;
    long long edgeCh = (long long)e * HID;

    // layer 1: H1 = x @ W1 ; A1 = scatter(H1) + b1  (ReLU fused into next GEMM load)
    k_gemm_wmma<128, false><<<blk(gemm_threads), T, 0, stream>>>(x, W1, Hbuf, n);
    k_agg_init <<<blk(nodeCh), T, 0, stream>>>(Hbuf, dinv, b1, Abuf, n);
    k_agg_edges<<<blk(edgeCh), T, 0, stream>>>(srcI, dstI, dinv, Hbuf, Abuf, e);

    // layer 2: H2 = relu(A1) @ W2 ; A2 = scatter(H2) + b2
    k_gemm_wmma<HID, true><<<blk(gemm_threads), T, 0, stream>>>(Abuf, W2, Hbuf, n);
    k_agg_init <<<blk(nodeCh), T, 0, stream>>>(Hbuf, dinv, b2, Abuf, n);
    k_agg_edges<<<blk(edgeCh), T, 0, stream>>>(srcI, dstI, dinv, Hbuf, Abuf, e);

    // layer 3: H3 = relu(A2) @ W3 ; A3 = scatter(H3) + b3 (no ReLU after)
    k_gemm_wmma<HID, true><<<blk(gemm_threads), T, 0, stream>>>(Abuf, W3, Hbuf, n);
    k_agg_init <<<blk(nodeCh), T, 0, stream>>>(Hbuf, dinv, b3, Abuf, n);
    k_agg_edges<<<blk(edgeCh), T, 0, stream>>>(srcI, dstI, dinv, Hbuf, Abuf, e);

    // global mean pool + FC
    k_pool_zero<<<blk((long long)G * HID), T, 0, stream>>>(P, cnt, G);
    k_pool     <<<blk(nodeCh), T, 0, stream>>>(Abuf, bat, P, cnt, n);
    k_fc       <<<blk((long long)G * NCLS), T, 0, stream>>>(P, cnt, Wfc, bfc, out, G, NCLS);
}